// BERTEncoder_6837587936132
// MI455X (gfx1250) — compile-verified
//
#include <hip/hip_runtime.h>
#include <stdint.h>

// Window-gather mean:  out[b,w,:] = mean(features[b, begin[b,w] : end[b,w], :])
// (window length clamped to MAX_WINDOW=8).
//
// MI455X plan (memory-bound: ~113 MFLOP vs ~150 MB of HBM traffic):
//  - features (100.7 MB) stay L2-resident (192 MB L2); output (50 MB) is
//    streamed with non-temporal b128 stores so it never evicts features
//  - each block stages its <=8x768 f32 tile into LDS with ONE Tensor Data
//    Mover descriptor (tensor_load_to_lds, TENSORcnt) issued by wave 0:
//    the DMA engine moves 24 KB while the vector pipe stays free
//  - all 6 wave32s then accumulate from LDS via ds_load_b128 (conflict-free
//    float4-per-lane layout), scale by 1/cnt, and NT-store the row

typedef float    v4f  __attribute__((ext_vector_type(4)));
typedef uint32_t u32x4 __attribute__((ext_vector_type(4)));
typedef uint32_t u32x8 __attribute__((ext_vector_type(8)));

namespace wm {
constexpr int B = 8;
constexpr int T = 4096;
constexpr int D = 768;
constexpr int W = 2048;
constexpr int MAXW = 8;
constexpr int CHUNKS = D / 4;          // 192 float4 chunks per row -> block size
}

__global__ __launch_bounds__(wm::CHUNKS) void window_mean_tdm_kernel(
    const float* __restrict__ feat,
    const int* __restrict__ begins,
    const int* __restrict__ ends,
    float* __restrict__ out)
{
    using namespace wm;

    __shared__ float smem[MAXW * D];            // 24 KB tile buffer

    const int gid = blockIdx.x;                 // window id in [0, B*W)
    const int b   = gid / W;                    // batch index (W pow2 -> shift)
    const int t   = threadIdx.x;                // float4 chunk id in [0, 192)

    // Block-uniform window descriptor (scalar loads).
    const int begin = begins[gid];
    const int end   = ends[gid];
    int cnt = end - begin;
    cnt = cnt < 0 ? 0 : (cnt > MAXW ? MAXW : cnt);

    // ---- Stage the cnt x 768 f32 tile into LDS with one TDM descriptor ----
    if (t < 32) {                               // wave 0 only issues the DMA
        const uint64_t gaddr =
            (uint64_t)(feat + ((size_t)b * T + (size_t)begin) * D); // tile start
        const uint32_t lds_base = (uint32_t)(uintptr_t)smem;        // LDS byte addr
        const uint32_t rows     = (uint32_t)(cnt < 1 ? 1 : cnt);

        // D# group 0 (128b): count=1 | lds_addr | global_addr[56:0] | type=2
        u32x4 g0 = {
            1u,                                                   // valid, no gather
            lds_base,                                             // bits [63:32]
            (uint32_t)gaddr,                                      // bits [95:64]
            ((uint32_t)(gaddr >> 32) & 0x01FFFFFFu) | (2u << 30)  // addr hi | type=2
        };
        // D# group 1 (256b): 2D tile, f32 elements, row stride = D elements
        u32x8 g1 = {
            (2u << 16),                 // workgroup_mask=0, data_size=2 (4 bytes)
            ((uint32_t)D << 16),        // abar_addr=0, tensor_dim0[15:0]=768
            (rows << 16),               // tensor_dim0 hi=0, tensor_dim1[15:0]=rows
            ((uint32_t)D << 16),        // tensor_dim1 hi=0, tile_dim0=768
            rows,                       // tile_dim1=rows, tile_dim2=0
            (uint32_t)D,                // tensor_dim0_stride[31:0]=768 elements
            0u,                         // stride0 hi=0, tensor_dim1_stride lo=0
            0u                          // tensor_dim1_stride hi=0
        };

        asm volatile("tensor_load_to_lds %0, %1"
                     :
                     : "s"(g0), "s"(g1)
                     : "memory");
        __builtin_amdgcn_s_wait_tensorcnt(0);   // tile landed in LDS
    }
    __syncthreads();                            // release the other 5 waves

    // ---- Accumulate from LDS (ds_load_b128, conflict-free layout) ---------
    v4f acc = {0.0f, 0.0f, 0.0f, 0.0f};
    const v4f* srow = (const v4f*)(&smem[t * 4]);
    for (int k = 0; k < cnt; ++k) {
        acc += srow[(size_t)k * CHUNKS];
    }

    const float inv = 1.0f / (float)(cnt < 1 ? 1 : cnt);
    acc *= inv;

    // ---- Streaming output: non-temporal b128 store (protect L2 residency) -
    v4f* dst = (v4f*)(out + (size_t)gid * D + (size_t)t * 4);
    __builtin_nontemporal_store(acc, dst);
}

extern "C" void kernel_launch(void* const* d_in, const int* in_sizes, int n_in,
                              void* d_out, int out_size, void* d_ws, size_t ws_size,
                              hipStream_t stream) {
    (void)in_sizes; (void)n_in; (void)out_size; (void)d_ws; (void)ws_size;

    const float* feat   = (const float*)d_in[0];
    const int*   begins = (const int*)d_in[1];
    const int*   ends   = (const int*)d_in[2];
    float*       out    = (float*)d_out;

    dim3 grid(wm::B * wm::W);       // 16384 blocks, one per window
    dim3 block(wm::CHUNKS);         // 192 threads = 6 wave32s

    hipLaunchKernelGGL(window_mean_tdm_kernel, grid, block, 0, stream,
                       feat, begins, ends, out);
}